// VID_NonLin_Layer_55301998903875
// MI455X (gfx1250) — compile-verified
//
#include <hip/hip_runtime.h>

// ---- problem constants (from reference) ----
#define STATE   300
#define CAUSE   40
#define NP      4
#define BATCH   256
#define TT      64
#define IMG     32
// flat element counts of the three outputs, concatenated in d_out
#define VID_ELEMS  (BATCH * IMG * IMG * TT)        // 16,777,216
#define PRED_ELEMS (BATCH * NP * STATE * TT)       // 19,660,800

typedef __attribute__((ext_vector_type(2))) float v2f;
typedef __attribute__((ext_vector_type(8))) float v8f;

// D = A(16x4) * B(4x16) + C  in fp32, CDNA5 WMMA
__device__ __forceinline__ v8f wmma4(v2f a, v2f b, v8f c) {
    return __builtin_amdgcn_wmma_f32_16x16x4_f32(
        /*neg_a=*/false, a, /*neg_b=*/false, b,
        /*c_mod=*/(short)0, c, /*reuse_a=*/false, /*reuse_b=*/false);
}

#define KC 20   // K-chunk staged in LDS (300 = 15 * 20, multiple of 4)

// ============================================================================
// Kernel 1: stacked GEMM  [A(300x300); C(256x300)] @ X(300 x 64) per (b,p).
//   grid.x = bp in [0,1024): one (b,p) pair, all 64 time steps (contiguous).
//   grid.y = 0..2 -> A rows [y*128, ...); 3..4 -> C rows [(y-3)*128, ...)
//   block  = 128 threads = 4 waves; wave w owns rows [w*32, w*32+32) x 64 cols
//   (2x4 WMMA tiles per K-step: 8 wmma fed by 2 A-frags + 4 B-frags).
// ============================================================================
__global__ __launch_bounds__(128)
void vidnl_main_gemm(const float* __restrict__ X,
                     const float* __restrict__ A,
                     const float* __restrict__ C,
                     float* __restrict__ out)
{
    __shared__ float Ws[128][KC];     // weight tile; stride 20 dwords is
                                      // conflict-free for b64 frag reads
    __shared__ v2f   Xs2[KC/2][80];   // X tile as k-pairs: {X[2k],X[2k+1]}[t]
                                      // stride 80 f2 -> halves hit disjoint banks

    const int tid  = threadIdx.x;
    const int lane = tid & 31;
    const int wv   = tid >> 5;          // 0..3
    const int lo   = lane & 15;
    const int hi   = lane >> 4;         // 0/1

    const int bp = blockIdx.x;          // b*4 + p
    const int mt = blockIdx.y;          // 0..4
    const bool isA = (mt < 3);
    const int m0 = isA ? mt * 128 : (mt - 3) * 128;
    const float* __restrict__ W = isA ? A : C;
    const int wrows = isA ? STATE : 256;

    v8f zero = {0.f,0.f,0.f,0.f,0.f,0.f,0.f,0.f};
    v8f acc[2][4] = {{zero, zero, zero, zero}, {zero, zero, zero, zero}};

    const int mrow0 = wv * 32 + lo;
    const int mrow1 = mrow0 + 16;

    for (int k0 = 0; k0 < STATE; k0 += KC) {
        // stage weight tile: 128 x 20 (2560 elems, 20/thread)
        #pragma unroll
        for (int r = 0; r < 20; ++r) {
            int i = tid + 128 * r;
            int row = i / KC, col = i % KC;
            int gr = m0 + row;
            Ws[row][col] = (gr < wrows) ? W[gr * STATE + (k0 + col)] : 0.0f;
        }
        // stage X tile as k-pairs: 10 x 64 float2 (640 elems, 5/thread)
        #pragma unroll
        for (int r = 0; r < 5; ++r) {
            int i = tid + 128 * r;              // 0..639
            int kk2 = i >> 6, t = i & 63;
            const float* src = X + (bp * STATE + (k0 + 2 * kk2)) * TT + t;
            v2f pr; pr.x = src[0]; pr.y = src[TT];
            Xs2[kk2][t] = pr;
        }
        // prefetch next chunk into cache while this one is consumed
        if (k0 + KC < STATE) {
            int gr = m0 + (tid >> 1);
            if (gr < wrows)
                __builtin_prefetch(&W[gr * STATE + (k0 + KC) + (tid & 1) * 16], 0, 3);
            __builtin_prefetch(&X[(bp * STATE + (k0 + KC) + (tid >> 6) * 10) * TT
                                  + (tid & 63)], 0, 3);
        }
        __syncthreads();

        #pragma unroll
        for (int ks = 0; ks < KC; ks += 4) {
            const int kp = (ks >> 1) + hi;      // pair-row for this lane half
            v2f a0 = ((const v2f*)&Ws[mrow0][0])[kp];
            v2f a1 = ((const v2f*)&Ws[mrow1][0])[kp];
            #pragma unroll
            for (int nt = 0; nt < 4; ++nt) {
                v2f b = Xs2[kp][nt * 16 + lo];
                acc[0][nt] = wmma4(a0, b, acc[0][nt]);
                acc[1][nt] = wmma4(a1, b, acc[1][nt]);
            }
        }
        __syncthreads();
    }

    const int b = bp >> 2, p = bp & 3;
    if (isA) {
        // X_pred[b,p,r,0,t+1] = (A @ X)[r, t]   (t == 63 dropped)
        float* __restrict__ pred = out + VID_ELEMS;
        #pragma unroll
        for (int mi = 0; mi < 2; ++mi) {
            #pragma unroll
            for (int nt = 0; nt < 4; ++nt) {
                int t = nt * 16 + lo;
                if (t < TT - 1) {
                    #pragma unroll
                    for (int v = 0; v < 8; ++v) {
                        int r = m0 + wv * 32 + mi * 16 + v + 8 * hi;
                        if (r < STATE)
                            pred[(bp * STATE + r) * TT + (t + 1)] = acc[mi][nt][v];
                    }
                }
            }
        }
    } else {
        // vid_recon: p=(gr,gc) 2x2 grid of 16x16 patches -> 32x32 image
        const int grr = p >> 1, gcc = p & 1;
        #pragma unroll
        for (int mi = 0; mi < 2; ++mi) {
            #pragma unroll
            for (int nt = 0; nt < 4; ++nt) {
                int t = nt * 16 + lo;
                #pragma unroll
                for (int v = 0; v < 8; ++v) {
                    int r = m0 + wv * 32 + mi * 16 + v + 8 * hi;  // pixel 0..255
                    int pr = r >> 4, pc = r & 15;
                    int row = grr * 16 + pr, col = gcc * 16 + pc;
                    out[((b * IMG + row) * IMG + col) * TT + t] = acc[mi][nt][v];
                }
            }
        }
    }
}

// ============================================================================
// Kernel 2: X_pred[...,0] = A @ x0.  N = 1024 columns (bp), K = M = 300.
//   grid.x = 0..15 (64 bp-columns each), grid.y = 0..2 (128 rows each).
//   block = 256 threads = 8 waves, 16M x 64N per wave.
// ============================================================================
__global__ __launch_bounds__(256)
void vidnl_x0_gemm(const float* __restrict__ x0,
                   const float* __restrict__ A,
                   float* __restrict__ out)
{
    __shared__ float Ws[128][KC];
    __shared__ v2f   Xs2[KC/2][80];

    const int tid  = threadIdx.x;
    const int lane = tid & 31;
    const int wv   = tid >> 5;
    const int lo   = lane & 15;
    const int hi   = lane >> 4;

    const int n0 = blockIdx.x * 64;
    const int m0 = blockIdx.y * 128;

    v8f zero = {0.f,0.f,0.f,0.f,0.f,0.f,0.f,0.f};
    v8f acc[4] = {zero, zero, zero, zero};
    const int mrow = wv * 16 + lo;

    for (int k0 = 0; k0 < STATE; k0 += KC) {
        #pragma unroll
        for (int r = 0; r < 10; ++r) {
            int i = tid + 256 * r;
            int row = i / KC, col = i % KC;
            int gr = m0 + row;
            Ws[row][col] = (gr < STATE) ? A[gr * STATE + (k0 + col)] : 0.0f;
        }
        #pragma unroll
        for (int r = 0; r < 3; ++r) {
            int i = tid + 256 * r;              // 0..639 (+guard)
            if (i < 640) {
                int kk2 = i >> 6, c = i & 63;
                // consecutive k are contiguous in x0 -> one aligned 8B load
                Xs2[kk2][c] = *(const v2f*)(x0 + (n0 + c) * STATE + k0 + 2 * kk2);
            }
        }
        __syncthreads();

        #pragma unroll
        for (int ks = 0; ks < KC; ks += 4) {
            const int kp = (ks >> 1) + hi;
            v2f a = ((const v2f*)&Ws[mrow][0])[kp];
            #pragma unroll
            for (int nt = 0; nt < 4; ++nt) {
                v2f b = Xs2[kp][nt * 16 + lo];
                acc[nt] = wmma4(a, b, acc[nt]);
            }
        }
        __syncthreads();
    }

    float* __restrict__ pred = out + VID_ELEMS;
    #pragma unroll
    for (int nt = 0; nt < 4; ++nt) {
        int bpcol = n0 + nt * 16 + lo;
        #pragma unroll
        for (int v = 0; v < 8; ++v) {
            int r = m0 + wv * 16 + v + 8 * hi;
            if (r < STATE)
                pred[(bpcol * STATE + r) * TT + 0] = acc[nt][v];
        }
    }
}

// ============================================================================
// Kernel 3: X_U = 0.1*(1+exp(-(Bm@U))) * sum_p |X|.  K = 40 fits in LDS.
//   grid.x = b in [0,256), grid.y = 0..2 (128 state rows each).
// ============================================================================
__global__ __launch_bounds__(256)
void vidnl_cause(const float* __restrict__ X,
                 const float* __restrict__ U,
                 const float* __restrict__ Bm,
                 float* __restrict__ out)
{
    __shared__ float Bs[128][44];      // stride 44: conflict-free b64 windows
    __shared__ v2f   Us2[CAUSE/2][80]; // U as k-pairs

    const int tid  = threadIdx.x;
    const int lane = tid & 31;
    const int wv   = tid >> 5;
    const int lo   = lane & 15;
    const int hi   = lane >> 4;

    const int b  = blockIdx.x;
    const int m0 = blockIdx.y * 128;

    // stage Bm tile: 128 x 40
    #pragma unroll
    for (int r = 0; r < 20; ++r) {
        int i = tid + 256 * r;                  // 0..5119
        int row = i / CAUSE, col = i % CAUSE;
        int gr = m0 + row;
        Bs[row][col] = (gr < STATE) ? Bm[gr * CAUSE + col] : 0.0f;
    }
    // stage U tile as k-pairs: 20 x 64 float2
    #pragma unroll
    for (int r = 0; r < 5; ++r) {
        int i = tid + 256 * r;                  // 0..1279
        int kk2 = i >> 6, t = i & 63;
        const float* src = U + (b * CAUSE + 2 * kk2) * TT + t;
        v2f pr; pr.x = src[0]; pr.y = src[TT];
        Us2[kk2][t] = pr;
    }
    __syncthreads();

    v8f zero = {0.f,0.f,0.f,0.f,0.f,0.f,0.f,0.f};
    v8f acc[4] = {zero, zero, zero, zero};
    const int mrow = wv * 16 + lo;

    #pragma unroll
    for (int ks = 0; ks < CAUSE; ks += 4) {
        const int kp = (ks >> 1) + hi;
        v2f a = ((const v2f*)&Bs[mrow][0])[kp];
        #pragma unroll
        for (int nt = 0; nt < 4; ++nt) {
            v2f bb = Us2[kp][nt * 16 + lo];
            acc[nt] = wmma4(a, bb, acc[nt]);
        }
    }

    float* __restrict__ xu = out + VID_ELEMS + PRED_ELEMS;
    #pragma unroll
    for (int nt = 0; nt < 4; ++nt) {
        int t = nt * 16 + lo;
        #pragma unroll
        for (int v = 0; v < 8; ++v) {
            int s = m0 + wv * 16 + v + 8 * hi;
            if (s < STATE) {
                float su = 0.0f;
                #pragma unroll
                for (int p = 0; p < NP; ++p)
                    su += fabsf(X[((b * NP + p) * STATE + s) * TT + t]);
                float g = 0.1f * (1.0f + expf(-acc[nt][v]));
                xu[(b * STATE + s) * TT + t] = g * su;
            }
        }
    }
}

// ============================================================================
// launch: inputs in setup_inputs() order: X, U, x0, A, Bm, C
// ============================================================================
extern "C" void kernel_launch(void* const* d_in, const int* in_sizes, int n_in,
                              void* d_out, int out_size, void* d_ws, size_t ws_size,
                              hipStream_t stream)
{
    (void)in_sizes; (void)n_in; (void)out_size; (void)d_ws; (void)ws_size;
    const float* X  = (const float*)d_in[0];
    const float* U  = (const float*)d_in[1];
    const float* x0 = (const float*)d_in[2];
    const float* A  = (const float*)d_in[3];
    const float* Bm = (const float*)d_in[4];
    const float* C  = (const float*)d_in[5];
    float* out = (float*)d_out;

    // stacked [A;C] @ X : recon + shifted X_pred
    vidnl_main_gemm<<<dim3(BATCH * NP, 5), 128, 0, stream>>>(X, A, C, out);
    // A @ x0 -> X_pred[..., 0]
    vidnl_x0_gemm<<<dim3(16, 3), 256, 0, stream>>>(x0, A, out);
    // cause path
    vidnl_cause<<<dim3(BATCH, 3), 256, 0, stream>>>(X, U, Bm, out);
}